// NumericalBiasModule_24481313587800
// MI455X (gfx1250) — compile-verified
//
#include <hip/hip_runtime.h>
#include <cstdint>

typedef float v2f __attribute__((ext_vector_type(2)));
typedef float v8f __attribute__((ext_vector_type(8)));

#define DIM 32
#define HID 64
#define LDS_STRIDE 130   // 128 + 2 pad: row stride mod 64 banks = 2 -> conflict-free b64 reads
#define EPS 1e-8f
#define LN2 0.69314718055994530942f

// One wave = one tile of 16 edges.
//   A (16x128, f32) : relation features, built per-lane directly in WMMA A-fragment layout
//   B (128x64, f32) : W1^T in padded LDS; fragments fetched with raw ds_load_b64 inline asm
//                     (asm volatile = not hoistable -> no VGPR blow-up / no spills), with
//                     software-pipelined s_wait_dscnt (DS ops retire in order).
//   C (16x64,  f32) : 4 accumulators of 16x16, f32 WMMA (full reference precision)
__global__ __launch_bounds__(256) void edge_bias_wmma_kernel(
    const float* __restrict__ x,
    const long long* __restrict__ ei,   // [2, E] int64
    const float* __restrict__ W1,       // [64, 128]
    const float* __restrict__ b1,       // [64]
    const float* __restrict__ W2,       // [1, 64]
    const float* __restrict__ b2,       // [1]
    float* __restrict__ out,            // [E]
    int E, int ntiles)
{
    __shared__ float sW1[HID * LDS_STRIDE];
    for (int i = threadIdx.x; i < HID * 128; i += blockDim.x)
        sW1[(i >> 7) * LDS_STRIDE + (i & 127)] = W1[i];
    __syncthreads();

    const int lane = threadIdx.x & 31;
    const int ln   = lane & 15;   // edge-in-tile (A rows) / output column n%16 (B cols)
    const int hi   = lane >> 4;   // K-half select within each 4-wide K step
    const int wave = (blockIdx.x * blockDim.x + threadIdx.x) >> 5;
    const int nwaves = (gridDim.x * blockDim.x) >> 5;

    // LDS byte address of this lane's B-fragment base:
    //   dword index (16t+ln)*130 + 4*(8f+j) + 2hi  ->  bytes: ln*520 + hi*8 (+ t*8320 + f*128 + j*16)
    const unsigned va0 = (unsigned)(uintptr_t)(&sW1[0]) + (unsigned)(ln * (LDS_STRIDE * 4) + hi * 8);

    // tiny layer-2 params: 4 values per lane, read once (L2/L0 resident)
    float b1v[4], w2v[4];
#pragma unroll
    for (int t = 0; t < 4; ++t) { b1v[t] = b1[16 * t + ln]; w2v[t] = W2[16 * t + ln]; }
    const float b2s = b2[0];

    for (int tile = wave; tile < ntiles; tile += nwaves) {
        const long long e0 = (long long)tile * 16;
        long long e = e0 + ln;
        if (e >= E) e = E - 1;                 // tail clamp (keeps EXEC all-ones)
        const long long src = ei[e];
        const long long dst = ei[(long long)E + e];
        const float* xi = x + src * DIM;
        const float* xj = x + dst * DIM;

        // Issue all gathers up front (16 x b64, L2-resident working set).
        // Lane owns dims d = 4j + {0,1} + 2*hi (j = 0..7).
        v2f xa[8], xb[8];
#pragma unroll
        for (int j = 0; j < 8; ++j) {
            xa[j] = *(const v2f*)(xi + 4 * j + 2 * hi);
            xb[j] = *(const v2f*)(xj + 4 * j + 2 * hi);
        }

        v8f acc[4] = {};
#pragma unroll
        for (int j = 0; j < 8; ++j) {
            // Kick off all 16 B-fragment loads for this j (f x t), waits pipelined below.
            v2f bf[4][4];
            const unsigned vaj = va0 + (unsigned)(16 * j);
            asm volatile(
                "ds_load_b64 %0,  %16 offset:0\n\t"
                "ds_load_b64 %1,  %16 offset:8320\n\t"
                "ds_load_b64 %2,  %16 offset:16640\n\t"
                "ds_load_b64 %3,  %16 offset:24960\n\t"
                "ds_load_b64 %4,  %16 offset:128\n\t"
                "ds_load_b64 %5,  %16 offset:8448\n\t"
                "ds_load_b64 %6,  %16 offset:16768\n\t"
                "ds_load_b64 %7,  %16 offset:25088\n\t"
                "ds_load_b64 %8,  %16 offset:256\n\t"
                "ds_load_b64 %9,  %16 offset:8576\n\t"
                "ds_load_b64 %10, %16 offset:16896\n\t"
                "ds_load_b64 %11, %16 offset:25216\n\t"
                "ds_load_b64 %12, %16 offset:384\n\t"
                "ds_load_b64 %13, %16 offset:8704\n\t"
                "ds_load_b64 %14, %16 offset:17024\n\t"
                "ds_load_b64 %15, %16 offset:25344"
                : "=v"(bf[0][0]), "=v"(bf[0][1]), "=v"(bf[0][2]), "=v"(bf[0][3]),
                  "=v"(bf[1][0]), "=v"(bf[1][1]), "=v"(bf[1][2]), "=v"(bf[1][3]),
                  "=v"(bf[2][0]), "=v"(bf[2][1]), "=v"(bf[2][2]), "=v"(bf[2][3]),
                  "=v"(bf[3][0]), "=v"(bf[3][1]), "=v"(bf[3][2]), "=v"(bf[3][3])
                : "v"(vaj));

            // 4 relation features for this dim pair (overlaps the LDS latency).
            v2f af[4];
#pragma unroll
            for (int o = 0; o < 2; ++o) {
                const float vi = xa[j][o], vj = xb[j][o];
                af[0][o] = vi * __builtin_amdgcn_rcpf(vj + EPS);                  // ratio
                af[1][o] = (__log2f(vi + EPS) - __log2f(vj + EPS)) * LN2;         // log_ratio
                const float ad = fabsf(vi - vj);
                af[2][o] = ad;                                                    // abs_diff
                af[3][o] = ad * __builtin_amdgcn_rcpf(fmaxf(vi, vj) + EPS);       // rel_diff
            }

            // DS retires in order: group f ready at DScnt <= 16-4(f+1).
            // The "+v" pass-through pins the consuming WMMAs after each wait.
#define WMMA_GROUP(F, WAITSTR)                                                            \
            asm volatile(WAITSTR : "+v"(bf[F][0]), "+v"(bf[F][1]),                        \
                                   "+v"(bf[F][2]), "+v"(bf[F][3]));                       \
            acc[0] = __builtin_amdgcn_wmma_f32_16x16x4_f32(                               \
                false, af[F], false, bf[F][0], (short)0, acc[0], false, false);           \
            acc[1] = __builtin_amdgcn_wmma_f32_16x16x4_f32(                               \
                false, af[F], false, bf[F][1], (short)0, acc[1], false, false);           \
            acc[2] = __builtin_amdgcn_wmma_f32_16x16x4_f32(                               \
                false, af[F], false, bf[F][2], (short)0, acc[2], false, false);           \
            acc[3] = __builtin_amdgcn_wmma_f32_16x16x4_f32(                               \
                false, af[F], false, bf[F][3], (short)0, acc[3], false, false)

            WMMA_GROUP(0, "s_wait_dscnt 0xc");
            WMMA_GROUP(1, "s_wait_dscnt 0x8");
            WMMA_GROUP(2, "s_wait_dscnt 0x4");
            WMMA_GROUP(3, "s_wait_dscnt 0x0");
#undef WMMA_GROUP
        }

        // Layer 2: h = relu(C + b1);  bias[m] = sum_n h[m][n] * W2[n] + b2.
        // C layout: VGPR r, lane -> row m = r + 8*hi, col n = 16t + ln.
        float p[8];
#pragma unroll
        for (int r = 0; r < 8; ++r) p[r] = 0.f;
#pragma unroll
        for (int t = 0; t < 4; ++t) {
#pragma unroll
            for (int r = 0; r < 8; ++r) {
                const float h = fmaxf(acc[t][r] + b1v[t], 0.f);
                p[r] = fmaf(h, w2v[t], p[r]);
            }
        }
        // reduce over the 16 lanes of each half (columns n)
#pragma unroll
        for (int m = 1; m <= 8; m <<= 1) {
#pragma unroll
            for (int r = 0; r < 8; ++r) p[r] += __shfl_xor(p[r], m, 32);
        }

        if (ln == 0) {
#pragma unroll
            for (int r = 0; r < 8; ++r) {
                const long long eo = e0 + 8 * hi + r;
                if (eo < E) out[eo] = p[r] + b2s;
            }
        }
    }
}

extern "C" void kernel_launch(void* const* d_in, const int* in_sizes, int n_in,
                              void* d_out, int out_size, void* d_ws, size_t ws_size,
                              hipStream_t stream) {
    const float*     x  = (const float*)d_in[0];
    const long long* ei = (const long long*)d_in[1];   // int64 edge_index [2, E]
    const float*     W1 = (const float*)d_in[2];
    const float*     b1 = (const float*)d_in[3];
    const float*     W2 = (const float*)d_in[4];
    const float*     b2 = (const float*)d_in[5];
    float*           out = (float*)d_out;

    const int E = in_sizes[1] / 2;
    const int ntiles = (E + 15) / 16;
    const int wavesPerBlock = 256 / 32;
    int blocks = (ntiles + wavesPerBlock - 1) / wavesPerBlock;
    if (blocks > 1024) blocks = 1024;   // grid-stride; amortizes the W1->LDS stage

    edge_bias_wmma_kernel<<<blocks, 256, 0, stream>>>(x, ei, W1, b1, W2, b2, out, E, ntiles);
}